// RelativePE_19224273617237
// MI455X (gfx1250) — compile-verified
//
#include <hip/hip_runtime.h>
#include <hip/hip_bf16.h>

typedef __attribute__((ext_vector_type(2))) float v2f;
typedef __attribute__((ext_vector_type(8))) float v8f;

#define WMMA_F32(a, b, c) \
  __builtin_amdgcn_wmma_f32_16x16x4_f32(false, (a), false, (b), (short)0, (c), false, false)

constexpr int Bc = 2;
constexpr int Sc = 4096;
constexpr int Dc = 64;     // head dim
constexpr int Rc = 128;    // radius
constexpr int Pc = 2 * Rc + 1;  // 257
constexpr int Gc = 8;      // t-split groups in pv kernel
constexpr long long ENERGY_ELEMS = (long long)Bc * Sc * Sc;
constexpr long long Z_ELEMS = (long long)Bc * Sc * Dc;

__device__ inline void atomic_add_f32(float* p, float v) {
  __hip_atomic_fetch_add(p, v, __ATOMIC_RELAXED, __HIP_MEMORY_SCOPE_AGENT);
}

// ---------------- Kernel 1: Q_rel[b,s,p] = sum_k Q[b,s,k] * E[p,k] ----------------
__global__ void __launch_bounds__(256) qrel_kernel(const float* __restrict__ Q,
                                                   const float* __restrict__ E,
                                                   float* __restrict__ qrel) {
  int idx = blockIdx.x * blockDim.x + threadIdx.x;
  const int total = Bc * Sc * Pc;
  if (idx >= total) return;
  int p  = idx % Pc;
  int bs = idx / Pc;
  const float* q = Q + (size_t)bs * Dc;
  const float* e = E + (size_t)p * Dc;
  float acc = 0.f;
#pragma unroll
  for (int k = 0; k < Dc; ++k) acc = fmaf(q[k], e[k], acc);
  qrel[idx] = acc;
}

// ---------------- Kernel 2: energy = (Q K^T + rel_shift) / 8, WMMA f32 ----------------
// One wave computes a 16(row) x 64(col) strip. Waves within a block share the K-tile.
__global__ void __launch_bounds__(256) energy_kernel(const float* __restrict__ Q,
                                                     const float* __restrict__ Km,
                                                     const float* __restrict__ qrel,
                                                     float* __restrict__ energy) {
  int wid  = (blockIdx.x * blockDim.x + threadIdx.x) >> 5;
  int lane = threadIdx.x & 31;
  int half = lane >> 4;
  int l15  = lane & 15;

  const int strips_per_b = (Sc / 16) * (Sc / 64);
  int b    = wid / strips_per_b;
  int rem  = wid % strips_per_b;
  int srow = rem % (Sc / 16);   // fastest within a block -> 8 waves share one K tile
  int t64  = rem / (Sc / 16);
  int s0   = srow * 16;

  // A-layout: lane (l15) = M row, VGPR j holds K = 4*kc + 2*half + j
  const float* qrow = Q + ((size_t)(b * Sc + s0 + l15)) * Dc + 2 * half;
  v2f a[16];
#pragma unroll
  for (int kc = 0; kc < 16; ++kc) a[kc] = *(const v2f*)(qrow + kc * 4);

  const float* qr = qrel + ((size_t)(b * Sc + s0)) * Pc;

#pragma unroll
  for (int tt = 0; tt < 4; ++tt) {
    int t0 = t64 * 64 + tt * 16;
    // B-layout: lane (l15) = N col; VGPR j holds K = 4*kc + 2*half + j
    const float* krow = Km + ((size_t)(b * Sc + t0 + l15)) * Dc + 2 * half;
    // Batch all B fragments first: one load clause, one wait, then WMMA chain.
    v2f bf[16];
#pragma unroll
    for (int kc = 0; kc < 16; ++kc) bf[kc] = *(const v2f*)(krow + kc * 4);

    v8f acc0 = {0.f, 0.f, 0.f, 0.f, 0.f, 0.f, 0.f, 0.f};
    v8f acc1 = {0.f, 0.f, 0.f, 0.f, 0.f, 0.f, 0.f, 0.f};
#pragma unroll
    for (int kc = 0; kc < 16; kc += 2) {   // two independent chains for ILP
      acc0 = WMMA_F32(a[kc],     bf[kc],     acc0);
      acc1 = WMMA_F32(a[kc + 1], bf[kc + 1], acc1);
    }
    v8f acc = acc0 + acc1;

    // D-layout epilogue: VGPR j, this lane -> row = j + 8*half, col = l15
#pragma unroll
    for (int j = 0; j < 8; ++j) {
      int row = j + 8 * half;
      int s   = s0 + row;
      int t   = t0 + l15;
      int rel = t - s;
      rel = rel < -Rc ? -Rc : (rel > Rc ? Rc : rel);
      float rs = qr[(size_t)row * Pc + (rel + Rc)];
      energy[((size_t)(b * Sc + s)) * Sc + t] = (acc[j] + rs) * 0.125f;
    }
  }
}

// ---------------- Kernel 3: per-row online (max, sum of exp) ----------------
__global__ void __launch_bounds__(256) rowstat_kernel(const float* __restrict__ energy,
                                                      float* __restrict__ rowmax,
                                                      float* __restrict__ rowsum) {
  int wid  = (blockIdx.x * blockDim.x + threadIdx.x) >> 5;
  int lane = threadIdx.x & 31;
  if (wid >= Bc * Sc) return;
  const float* row = energy + (size_t)wid * Sc;
  float m = -3.0e38f, ssum = 0.f;
  for (int i = lane; i < Sc; i += 32) {
    float e  = row[i];
    float mn = fmaxf(m, e);
    ssum = ssum * __expf(m - mn) + __expf(e - mn);
    m = mn;
  }
#pragma unroll
  for (int off = 16; off > 0; off >>= 1) {
    float m2 = __shfl_xor(m, off, 32);
    float s2 = __shfl_xor(ssum, off, 32);
    float mn = fmaxf(m, m2);
    ssum = ssum * __expf(m - mn) + s2 * __expf(m2 - mn);
    m = mn;
  }
  if (lane == 0) { rowmax[wid] = m; rowsum[wid] = ssum; }
}

// ---------------- Kernel 4: Z = softmax(energy) @ V via WMMA, split over t ----------------
// Waves within a block share g (same V rows) -> V reads hit the WGP cache.
__global__ void __launch_bounds__(256) pv_kernel(const float* __restrict__ energy,
                                                 const float* __restrict__ V,
                                                 const float* __restrict__ rowmax,
                                                 const float* __restrict__ rowsum,
                                                 float* __restrict__ zout) {
  int wid  = (blockIdx.x * blockDim.x + threadIdx.x) >> 5;
  int lane = threadIdx.x & 31;
  int half = lane >> 4;
  int l15  = lane & 15;

  const int per_b = (Sc / 16) * Gc;
  int b    = wid / per_b;
  int rem  = wid % per_b;
  int sblk = rem % (Sc / 16);   // fastest within a block -> shared g / V rows
  int g    = rem / (Sc / 16);
  int s0   = sblk * 16;

  float rm = rowmax[b * Sc + s0 + l15];  // all 8 A-values in this lane share row l15

  v8f acc[4];
#pragma unroll
  for (int nt = 0; nt < 4; ++nt) acc[nt] = (v8f){0.f,0.f,0.f,0.f,0.f,0.f,0.f,0.f};

  const int chunks = (Sc / 16) / Gc;  // 32 chunks of 16 columns
  const float* erow = energy + ((size_t)(b * Sc + s0 + l15)) * Sc + 2 * half;

  for (int ci = 0; ci < chunks; ++ci) {
    int t0 = (g * chunks + ci) * 16;

    // Load E chunk (A-layout access) and all V fragments (B-layout) up front.
    v2f ev[4];
#pragma unroll
    for (int cc = 0; cc < 4; ++cc) ev[cc] = *(const v2f*)(erow + t0 + cc * 4);

    const float* vbase = V + ((size_t)(b * Sc + t0 + 2 * half)) * Dc + l15;
    v2f bv[4][4];
#pragma unroll
    for (int nt = 0; nt < 4; ++nt) {
#pragma unroll
      for (int kc = 0; kc < 4; ++kc) {
        const float* vp = vbase + (size_t)(kc * 4) * Dc + nt * 16;
        v2f t;
        t.x = vp[0];
        t.y = vp[Dc];
        bv[nt][kc] = t;
      }
    }

    v2f pa[4];
#pragma unroll
    for (int cc = 0; cc < 4; ++cc) {
      v2f p;
      p.x = __expf(ev[cc].x - rm);
      p.y = __expf(ev[cc].y - rm);
      pa[cc] = p;
    }

#pragma unroll
    for (int kc = 0; kc < 4; ++kc) {   // 4 independent nt chains per kc step
#pragma unroll
      for (int nt = 0; nt < 4; ++nt) {
        acc[nt] = WMMA_F32(pa[kc], bv[nt][kc], acc[nt]);
      }
    }
  }

  float inv[8];
#pragma unroll
  for (int j = 0; j < 8; ++j) {
    int row = j + 8 * half;
    inv[j] = 1.0f / rowsum[b * Sc + s0 + row];
  }
#pragma unroll
  for (int nt = 0; nt < 4; ++nt) {
#pragma unroll
    for (int j = 0; j < 8; ++j) {
      int row = j + 8 * half;
      size_t zi = ((size_t)(b * Sc + s0 + row)) * Dc + nt * 16 + l15;
      atomic_add_f32(&zout[zi], acc[nt][j] * inv[j]);
    }
  }
}

extern "C" void kernel_launch(void* const* d_in, const int* in_sizes, int n_in,
                              void* d_out, int out_size, void* d_ws, size_t ws_size,
                              hipStream_t stream) {
  const float* Q  = (const float*)d_in[0];
  const float* K  = (const float*)d_in[1];
  const float* V  = (const float*)d_in[2];
  const float* RE = (const float*)d_in[3];
  // d_in[4] = segment_ids (unused: segmented=False in reference)

  float* energy = (float*)d_out;
  float* zout   = (float*)d_out + ENERGY_ELEMS;

  float* qrel   = (float*)d_ws;                       // B*S*P floats (8.4 MB)
  float* rowmax = qrel + (size_t)Bc * Sc * Pc;        // B*S floats
  float* rowsum = rowmax + (size_t)Bc * Sc;           // B*S floats

  // K1: Q_rel
  {
    int total  = Bc * Sc * Pc;
    int blocks = (total + 255) / 256;
    qrel_kernel<<<blocks, 256, 0, stream>>>(Q, RE, qrel);
  }
  // K2: energy (WMMA f32)
  {
    int waves = Bc * (Sc / 16) * (Sc / 64);  // 32768
    energy_kernel<<<waves / 8, 256, 0, stream>>>(Q, K, qrel, energy);
  }
  // Zero the Z accumulation region (atomics accumulate into it)
  hipMemsetAsync(zout, 0, (size_t)Z_ELEMS * sizeof(float), stream);
  // K3: row stats
  {
    int waves = Bc * Sc;  // 8192
    rowstat_kernel<<<waves / 8, 256, 0, stream>>>(energy, rowmax, rowsum);
  }
  // K4: Z = softmax @ V (WMMA f32, t-split with atomic accumulation)
  {
    int waves = Bc * (Sc / 16) * Gc;  // 4096
    pv_kernel<<<waves / 8, 256, 0, stream>>>(energy, V, rowmax, rowsum, zout);
  }
}